// ResGraphConvUnpool_85487029059749
// MI455X (gfx1250) — compile-verified
//
#include <hip/hip_runtime.h>
#include <hip/hip_bf16.h>

typedef __attribute__((ext_vector_type(2))) float v2f;
typedef __attribute__((ext_vector_type(8))) float v8f;

#define NB 16
#define NPTS 2048
#define CCH 128
#define NROWS (NB * NPTS)          // 32768
#define KNN 7                      // neighbors excluding self
#define MT 4                       // m-tiles per block in the GCN GEMM
#define FLT_BIG 3.402823466e+38f

// ---------------------------------------------------------------------------
// Tiled transpose: src (B, R, Cn) -> dst (B, Cn, R). grid (Cn/32, R/32, B), 256 thr
// ---------------------------------------------------------------------------
__global__ __launch_bounds__(256) void transpose_kernel(
    const float* __restrict__ src, float* __restrict__ dst, int R, int Cn) {
  __shared__ float tile[32][33];
  const float* s = src + (size_t)blockIdx.z * R * Cn;
  float* d = dst + (size_t)blockIdx.z * R * Cn;
  const int c0 = blockIdx.x * 32, r0 = blockIdx.y * 32;
  const int tc = threadIdx.x & 31, tr = threadIdx.x >> 5;  // tr in 0..7
#pragma unroll
  for (int i = 0; i < 4; ++i) {
    int rr = tr + i * 8;
    tile[rr][tc] = s[(size_t)(r0 + rr) * Cn + (c0 + tc)];
  }
  __syncthreads();
#pragma unroll
  for (int i = 0; i < 4; ++i) {
    int rr = tr + i * 8;
    d[(size_t)(c0 + rr) * R + (r0 + tc)] = tile[tc][rr];
  }
}

// ---------------------------------------------------------------------------
// KNN: 7 nearest neighbors (excluding self) per point. Row sums are always 8,
// so normalized adjacency == adj/8; we only need indices (stored as global rows).
// grid = B*(N/256) = 128 blocks, 256 threads.
// ---------------------------------------------------------------------------
__global__ __launch_bounds__(256) void knn_kernel(
    const float* __restrict__ xyz, int* __restrict__ nbr) {
  __shared__ float sx[256], sy[256], sz[256];
  const int b = blockIdx.x >> 3;
  const int n = ((blockIdx.x & 7) << 8) + threadIdx.x;
  const float* xb = xyz + (size_t)b * 3 * NPTS;
  const float qx = xb[n], qy = xb[NPTS + n], qz = xb[2 * NPTS + n];
  const float q2 = qx * qx + qy * qy + qz * qz;
  float bestd[KNN];
  int besti[KNN];
#pragma unroll
  for (int k = 0; k < KNN; ++k) { bestd[k] = FLT_BIG; besti[k] = 0; }
  for (int t = 0; t < NPTS / 256; ++t) {
    __syncthreads();
    const int jj0 = t * 256;
    sx[threadIdx.x] = xb[jj0 + threadIdx.x];
    sy[threadIdx.x] = xb[NPTS + jj0 + threadIdx.x];
    sz[threadIdx.x] = xb[2 * NPTS + jj0 + threadIdx.x];
    __syncthreads();
    for (int j = 0; j < 256; ++j) {
      const int jj = jj0 + j;
      if (jj == n) continue;
      const float cx = sx[j], cy = sy[j], cz = sz[j];
      const float c2 = cx * cx + cy * cy + cz * cz;
      const float d = q2 + c2 - 2.0f * (qx * cx + qy * cy + qz * cz);
      if (d < bestd[KNN - 1]) {
        int k = KNN - 1;
        while (k > 0 && d < bestd[k - 1]) {
          bestd[k] = bestd[k - 1]; besti[k] = besti[k - 1]; --k;
        }
        bestd[k] = d; besti[k] = jj;
      }
    }
  }
  const int m = b * NPTS + n;
#pragma unroll
  for (int k = 0; k < KNN; ++k) nbr[m * KNN + k] = b * NPTS + besti[k];
}

// ---------------------------------------------------------------------------
// GCN block dual GEMM (f32 WMMA 16x16x4):
//   ptsOut[m,:] = ptsIn[m,:] + (8/9) * relu(ptsIn) @ Wc^T   (center path)
//   Yg[m,:]     = ptsIn @ Wg^T                               (pre-gather path)
// One block = 64 rows (4 m-tiles) staged in LDS; 8 waves cover the 8 column
// tiles. Each weight-fragment load feeds 4 A-fragments -> 8 WMMAs, cutting L2
// weight traffic 4x vs one-m-tile blocks. grid = NROWS/64 = 512 blocks.
// ---------------------------------------------------------------------------
__global__ __launch_bounds__(256) void gcn_gemm_kernel(
    const float* __restrict__ ptsIn, const float* __restrict__ Wc,
    const float* __restrict__ Wg, float* __restrict__ ptsOut,
    float* __restrict__ Yg) {
  __shared__ float ldsX[MT * 16 * CCH];  // 32 KB
  const int tid = threadIdx.x;
  const int m0 = blockIdx.x * (MT * 16);
  {
    const float4* s4 = (const float4*)(ptsIn + (size_t)m0 * CCH);
    float4* d4 = (float4*)ldsX;
#pragma unroll
    for (int i = 0; i < 2 * MT; ++i) d4[tid + i * 256] = s4[tid + i * 256];
  }
  const int wave = tid >> 5;       // n_tile 0..7
  const int lane = tid & 31;
  const int half = lane >> 4;      // 0: K pair (0,1); 1: K pair (2,3)
  const int r = lane & 15;         // A row / B column within tile
  const int dcol = wave * 16 + r;  // weight row == output column
  // pre-warm this wave's weight rows (global_prefetch_b8)
  __builtin_prefetch(Wc + (size_t)dcol * CCH, 0, 3);
  __builtin_prefetch(Wg + (size_t)dcol * CCH, 0, 3);
  __syncthreads();
  v8f acc_c[MT] = {};
  v8f acc_g[MT] = {};
  for (int c = 0; c < CCH; c += 4) {
    const int kb = c + 2 * half;
    const v2f bc = *(const v2f*)(Wc + (size_t)dcol * CCH + kb);
    const v2f bg = *(const v2f*)(Wg + (size_t)dcol * CCH + kb);
#pragma unroll
    for (int mt = 0; mt < MT; ++mt) {
      const float* ax = ldsX + (mt * 16 + r) * CCH + kb;
      const float a0 = ax[0], a1 = ax[1];
      v2f ag = {a0, a1};
      v2f ac = {fmaxf(a0, 0.0f), fmaxf(a1, 0.0f)};
      acc_c[mt] = __builtin_amdgcn_wmma_f32_16x16x4_f32(
          false, ac, false, bc, (short)0, acc_c[mt], false, false);
      acc_g[mt] = __builtin_amdgcn_wmma_f32_16x16x4_f32(
          false, ag, false, bg, (short)0, acc_g[mt], false, false);
    }
  }
  // D layout: vgpr v, lane L -> m = v + 8*(L/16), n = L%16
#pragma unroll
  for (int mt = 0; mt < MT; ++mt) {
#pragma unroll
    for (int v = 0; v < 8; ++v) {
      const int m_loc = mt * 16 + v + 8 * half;
      const size_t gidx = (size_t)(m0 + m_loc) * CCH + dcol;
      ptsOut[gidx] = ldsX[m_loc * CCH + dcol] + (8.0f / 9.0f) * acc_c[mt][v];
      Yg[gidx] = acc_g[mt][v];
    }
  }
}

// ---------------------------------------------------------------------------
// Gather-average of Yg over {self + 7 neighbors}, scaled by (1-alpha)/8 = 1/72,
// accumulated into ptsOut. float4 per thread; 32 consecutive threads share one
// row -> every neighbor-row read is a fully coalesced 512B line.
// grid = NROWS*32/256 = 4096 blocks.
// ---------------------------------------------------------------------------
__global__ __launch_bounds__(256) void gcn_gather_kernel(
    const float* __restrict__ Yg, const int* __restrict__ nbr,
    float* __restrict__ ptsOut) {
  const int idx = blockIdx.x * 256 + threadIdx.x;  // < 32768*32
  const int m = idx >> 5;
  const int q = idx & 31;  // float4 slot within the 128-ch row
  const float4* yg4 = (const float4*)Yg;
  float4 s = yg4[(size_t)m * 32 + q];
  const int* nb = nbr + m * KNN;
#pragma unroll
  for (int j = 0; j < KNN; ++j) {
    const float4 t = yg4[(size_t)nb[j] * 32 + q];
    s.x += t.x; s.y += t.y; s.z += t.z; s.w += t.w;
  }
  float4* po4 = (float4*)ptsOut;
  float4 p = po4[idx];
  p.x += (1.0f / 72.0f) * s.x;
  p.y += (1.0f / 72.0f) * s.y;
  p.z += (1.0f / 72.0f) * s.z;
  p.w += (1.0f / 72.0f) * s.w;
  po4[idx] = p;
}

// ---------------------------------------------------------------------------
// Final unpool GEMM: U[m, 0:6] = pts @ Wuc^T, U[m, 6:12] = pts @ Wug^T.
// 12 output columns packed into one 16-wide WMMA tile (lanes 12..15 zero).
// grid = 256 blocks x 8 waves = 2048 m-tiles.
// ---------------------------------------------------------------------------
__global__ __launch_bounds__(256) void final_gemm_kernel(
    const float* __restrict__ pts, const float* __restrict__ Wuc,
    const float* __restrict__ Wug, float* __restrict__ U) {
  const int wave = threadIdx.x >> 5;
  const int lane = threadIdx.x & 31;
  const int m_tile = blockIdx.x * 8 + wave;
  const int half = lane >> 4;
  const int r = lane & 15;
  v8f acc = {};
#pragma unroll 4
  for (int c = 0; c < CCH; c += 4) {
    const int kb = c + 2 * half;
    v2f a = *(const v2f*)(pts + (size_t)(m_tile * 16 + r) * CCH + kb);
    v2f b;
    if (r < 6)        b = *(const v2f*)(Wuc + r * CCH + kb);
    else if (r < 12)  b = *(const v2f*)(Wug + (r - 6) * CCH + kb);
    else              b = (v2f){0.0f, 0.0f};
    acc = __builtin_amdgcn_wmma_f32_16x16x4_f32(false, a, false, b,
                                                (short)0, acc, false, false);
  }
#pragma unroll
  for (int v = 0; v < 8; ++v) {
    const int m_loc = v + 8 * half;
    if (r < 12) U[(size_t)(m_tile * 16 + m_loc) * 12 + r] = acc[v];
  }
}

// ---------------------------------------------------------------------------
// Final combine: new = (8/9)*Uc + (1/72)*Sum8(Ug), interleave + add xyz,
// write new_xyz (B,3,2N). grid = 768 blocks, 256 thr (one per (m,o)).
// ---------------------------------------------------------------------------
__global__ __launch_bounds__(256) void final_out_kernel(
    const float* __restrict__ U, const int* __restrict__ nbr,
    const float* __restrict__ xyz, float* __restrict__ out) {
  const int idx = blockIdx.x * 256 + threadIdx.x;  // < 32768*6
  const int m = idx / 6;
  const int o = idx - m * 6;
  float uc = U[m * 12 + o];
  float ug = U[m * 12 + 6 + o];
  const int* nb = nbr + m * KNN;
#pragma unroll
  for (int j = 0; j < KNN; ++j) ug += U[nb[j] * 12 + 6 + o];
  const float val = (8.0f / 9.0f) * uc + (1.0f / 72.0f) * ug;
  const int b = m >> 11;
  const int n = m & (NPTS - 1);
  const int cc = o >> 1, s = o & 1;
  out[(size_t)(b * 3 + cc) * (2 * NPTS) + s * NPTS + n] =
      val + xyz[(size_t)(b * 3 + cc) * NPTS + n];
}

// ---------------------------------------------------------------------------
extern "C" void kernel_launch(void* const* d_in, const int* in_sizes, int n_in,
                              void* d_out, int out_size, void* d_ws, size_t ws_size,
                              hipStream_t stream) {
  const float* xyz    = (const float*)d_in[0];  // (16,3,2048)
  const float* points = (const float*)d_in[1];  // (16,128,2048)
  const float* Wc     = (const float*)d_in[2];  // (4,128,128)
  const float* Wg     = (const float*)d_in[3];  // (4,128,128)
  const float* Wuc    = (const float*)d_in[4];  // (6,128)
  const float* Wug    = (const float*)d_in[5];  // (6,128)
  float* out = (float*)d_out;

  char* ws = (char*)d_ws;
  int*   nbr  = (int*)ws;                               // 32768*7 ints
  float* ptsA = (float*)(ws + (size_t)1  * (1 << 20));  // 16 MB
  float* ptsB = (float*)(ws + (size_t)17 * (1 << 20));  // 16 MB
  float* Yg   = (float*)(ws + (size_t)33 * (1 << 20));  // 16 MB
  float* U    = (float*)(ws + (size_t)49 * (1 << 20));  // 32768*12 floats

  // points (B,C,N) -> pts (B,N,C)
  transpose_kernel<<<dim3(64, 4, NB), 256, 0, stream>>>(points, ptsA, CCH, NPTS);
  // neighbor indices (normalized adjacency == adj/8; only indices needed)
  knn_kernel<<<128, 256, 0, stream>>>(xyz, nbr);

  float* pin = ptsA;
  float* pout = ptsB;
  for (int i = 0; i < 4; ++i) {
    gcn_gemm_kernel<<<NROWS / (MT * 16), 256, 0, stream>>>(
        pin, Wc + (size_t)i * CCH * CCH, Wg + (size_t)i * CCH * CCH, pout, Yg);
    gcn_gather_kernel<<<NROWS * (CCH / 4) / 256, 256, 0, stream>>>(Yg, nbr, pout);
    float* t = pin; pin = pout; pout = t;
  }
  // after 4 swaps, final pts lives in ptsA (== pin)
  final_gemm_kernel<<<NROWS / 16 / 8, 256, 0, stream>>>(pin, Wuc, Wug, U);
  final_out_kernel<<<NROWS * 6 / 256, 256, 0, stream>>>(U, nbr, xyz, out);
  // pts (B,N,C) -> out2 (B,C,N), appended after new_xyz (16*3*4096 floats)
  transpose_kernel<<<dim3(4, 64, NB), 256, 0, stream>>>(
      pin, out + (size_t)NB * 3 * 2 * NPTS, NPTS, CCH);
}